// KalmanFilter_18056042512874
// MI455X (gfx1250) — compile-verified
//
#include <hip/hip_runtime.h>
#include <stdint.h>

#define B_ 4096
#define T_ 200
#define S_ 8
#define M_ 2

#define BPB 32            // batches per block (8 lanes per batch * 32 = 256 threads)
#define THREADS 256
#define NBLOCKS (B_ / BPB)

// compiler-only reordering fence (HW keeps per-wave DS/TDM issue order sane)
#define LDSFENCE() asm volatile("" ::: "memory")

#if defined(__AMDGCN__) && __has_builtin(__builtin_amdgcn_tensor_load_to_lds) && \
    __has_builtin(__builtin_amdgcn_s_wait_tensorcnt)
#define HAVE_TDM 1
#else
#define HAVE_TDM 0
#endif

typedef uint32_t v4u __attribute__((ext_vector_type(4)));
typedef int      v8i __attribute__((ext_vector_type(8)));
typedef int      v4i __attribute__((ext_vector_type(4)));

__device__ __forceinline__ uint32_t lds_off(const void* p) {
    // generic LDS pointer = {shared aperture, 32-bit LDS byte offset}
    return (uint32_t)(uintptr_t)p;
}

__device__ __forceinline__ int rfl(int x) {
#if defined(__HIP_DEVICE_COMPILE__) && defined(__AMDGCN__)
    return __builtin_amdgcn_readfirstlane(x);
#else
    return x;
#endif
}

#if HAVE_TDM
// 2-D tile load: `rows` rows of `d0` fp32 elements, row stride in elements.
__device__ __forceinline__ void tdm_load_2d(uint32_t ldsAddr, uint64_t gaddr,
                                            uint32_t d0, uint32_t rows,
                                            uint64_t strideElems) {
    v4u g0;
    g0.x = 1u;                                   // count=1, gather off
    g0.y = ldsAddr;                              // lds_addr
    g0.z = (uint32_t)(gaddr & 0xFFFFFFFFull);    // global_addr[31:0]
    g0.w = (uint32_t)((gaddr >> 32) & 0x1FFFFFFull) | 0x80000000u; // addr[56:32] | type=2
    v8i g1;
    g1[0] = (int)(2u << 16);                                 // data_size = 4 bytes
    g1[1] = (int)((d0 & 0xFFFFu) << 16);                     // tensor_dim0[15:0]
    g1[2] = (int)((d0 >> 16) | ((rows & 0xFFFFu) << 16));    // td0[31:16] | tensor_dim1[15:0]
    g1[3] = (int)((rows >> 16) | ((d0 & 0xFFFFu) << 16));    // td1[31:16] | tile_dim0
    g1[4] = (int)(rows & 0xFFFFu);                           // tile_dim1 (tile_dim2 = 0)
    g1[5] = (int)(strideElems & 0xFFFFFFFFull);              // tensor_dim0_stride[31:0]
    g1[6] = (int)((strideElems >> 32) & 0xFFFFull);          // stride[47:32]
    g1[7] = 0;
    v4i g2 = {0, 0, 0, 0};
    v4i g3 = {0, 0, 0, 0};
    v8i g4 = {0, 0, 0, 0, 0, 0, 0, 0};           // extra group (6-arg clang-23 form)
    __builtin_amdgcn_tensor_load_to_lds(g0, g1, g2, g3, g4, 0);
}
#endif

__global__ __launch_bounds__(THREADS)
void kalman1250_kernel(const float* __restrict__ Yg, const float* __restrict__ Fg,
                       const float* __restrict__ Hg, const float* __restrict__ Qg,
                       const float* __restrict__ Rg, const float* __restrict__ im_g,
                       const float* __restrict__ ic_g, float* __restrict__ means,
                       float* __restrict__ covs)
{
    __shared__ __align__(16) float sF[2][BPB][64];
    __shared__ __align__(16) float sH[2][BPB][16];
    __shared__ __align__(16) float sY[2][BPB * 2];
    __shared__ __align__(16) float sX[BPB][80];   // per-group exchange scratch

    const int tid = threadIdx.x;
    const int lb  = tid >> 3;        // local batch 0..31
    const int j   = tid & 7;         // owned column / row index
    const int b0  = blockIdx.x * BPB;
    const int b   = b0 + lb;

    // ---- persistent per-lane state ----
    float pcol[8], meanr[8], qcol[8], idrow[8];
    #pragma unroll
    for (int k = 0; k < 8; ++k) {
        pcol[k]  = ic_g[k * 8 + j];       // init_cov column j (== row j, symmetric)
        meanr[k] = im_g[b * 8 + k];       // replicated mean
        qcol[k]  = Qg[k * 8 + j];
        idrow[k] = (k == j) ? 1.0f : 0.0f;
    }
    float mmine = im_g[b * 8 + j];
    const float R00 = Rg[0], R01 = Rg[1], R11 = Rg[3];

    // ---- t = 0 outputs ----
    means[(size_t)b * T_ * S_ + j] = mmine;
    {
        float4* cd = (float4*)&covs[((size_t)b * T_) * 64 + j * 8];
        cd[0] = make_float4(pcol[0], pcol[1], pcol[2], pcol[3]);
        cd[1] = make_float4(pcol[4], pcol[5], pcol[6], pcol[7]);
    }

#if HAVE_TDM
    // Each wave stages only its own 4 batches: barrier-free double buffering.
    // All descriptor inputs are made wave-uniform scalars up front.
    const int wvu = rfl(tid >> 5);                 // wave id (0..7), SGPR
    const uint32_t ldsF0 = lds_off(&sF[0][wvu * 4][0]);
    const uint32_t ldsF1 = lds_off(&sF[1][wvu * 4][0]);
    const uint32_t ldsH0 = lds_off(&sH[0][wvu * 4][0]);
    const uint32_t ldsH1 = lds_off(&sH[1][wvu * 4][0]);
    const uint32_t ldsY0 = lds_off(&sY[0][wvu * 8]);
    const uint32_t ldsY1 = lds_off(&sY[1][wvu * 8]);
    const uint64_t rowW = (uint64_t)(b0 + wvu * 4) * (uint64_t)T_;   // first row of this wave
    const uint64_t gF = (uint64_t)(uintptr_t)Fg + rowW * 256ull;     // 64 floats/row
    const uint64_t gH = (uint64_t)(uintptr_t)Hg + rowW * 64ull;      // 16 floats/row
    const uint64_t gY = (uint64_t)(uintptr_t)Yg + rowW * 8ull;       //  2 floats/row
    auto issue = [&](int tu, int buf) {
        const uint64_t tb = (uint64_t)(uint32_t)tu;
        tdm_load_2d(buf ? ldsF1 : ldsF0, gF + tb * 256ull, 64u, 4u, (uint64_t)T_ * 64ull);
        tdm_load_2d(buf ? ldsH1 : ldsH0, gH + tb * 64ull,  16u, 4u, (uint64_t)T_ * 16ull);
        tdm_load_2d(buf ? ldsY1 : ldsY0, gY + tb * 8ull,    2u, 4u, (uint64_t)T_ * 2ull);
    };
    issue(0, 0);
#else
    // Fallback: per-lane cooperative copy of this wave's batches (in-order DS).
    auto stage = [&](int tu, int buf) {
        const size_t row = ((size_t)(b0 + lb)) * T_ + tu;
        const float4* fs = (const float4*)&Fg[row * 64 + j * 8];
        float4* fd = (float4*)&sF[buf][lb][j * 8];
        fd[0] = fs[0]; fd[1] = fs[1];
        *(float2*)&sH[buf][lb][j * 2] = *(const float2*)&Hg[row * 16 + j * 2];
        if (j < 2) sY[buf][lb * 2 + j] = Yg[row * 2 + j];
    };
    stage(0, 0);
#endif

    for (int t = 0; t <= T_ - 2; ++t) {
        {
            int tn = (t + 1 <= T_ - 2) ? (t + 1) : (T_ - 2);
            LDSFENCE();
#if HAVE_TDM
            issue(tn, (t + 1) & 1);
            __builtin_amdgcn_s_wait_tensorcnt(3);  // this wave's tiles for step t done
#else
            stage(tn, (t + 1) & 1);
#endif
            LDSFENCE();
        }
        const int buf = t & 1;

        const float* fB = &sF[buf][lb][0];
        const float* hB = &sH[buf][lb][0];
        const float4* f4 = (const float4*)fB;
        const float y0 = sY[buf][lb * 2 + 0];
        const float y1 = sY[buf][lb * 2 + 1];

        const float4 ha = ((const float4*)hB)[0], hb4 = ((const float4*)hB)[1];
        const float4 hcv = ((const float4*)hB)[2], hd4 = ((const float4*)hB)[3];
        const float h0[8] = {ha.x, ha.y, ha.z, ha.w, hb4.x, hb4.y, hb4.z, hb4.w};
        const float h1[8] = {hcv.x, hcv.y, hcv.z, hcv.w, hd4.x, hd4.y, hd4.z, hd4.w};
        const float hc0 = hB[j], hc1 = hB[8 + j];
        const float4 fra = f4[j * 2], frb = f4[j * 2 + 1];
        const float Frow[8] = {fra.x, fra.y, fra.z, fra.w, frb.x, frb.y, frb.z, frb.w};

        // 1) residual (replicated)
        float hm0 = 0.f, hm1 = 0.f;
        #pragma unroll
        for (int k = 0; k < 8; ++k) { hm0 += h0[k] * meanr[k]; hm1 += h1[k] * meanr[k]; }
        const float r0 = y0 - hm0, r1 = y1 - hm1;

        // 2) HP column j
        float hp0 = 0.f, hp1 = 0.f;
        #pragma unroll
        for (int k = 0; k < 8; ++k) { hp0 += h0[k] * pcol[k]; hp1 += h1[k] * pcol[k]; }

        // 3) Smat = HP H^T + R : butterfly reduce over the 8 lanes of the group
        float p00 = hp0 * hc0, p01 = hp0 * hc1, p11 = hp1 * hc1;
        #pragma unroll
        for (int m = 1; m <= 4; m <<= 1) {
            p00 += __shfl_xor(p00, m, 32);
            p01 += __shfl_xor(p01, m, 32);
            p11 += __shfl_xor(p11, m, 32);
        }
        const float s00 = p00 + R00, s01 = p01 + R01, s11 = p11 + R11;
        const float rdet = 1.0f / (s00 * s11 - s01 * s01);
        const float si00 = s11 * rdet, si01 = -s01 * rdet, si11 = s00 * rdet;

        // 5) K row j
        const float k0 = si00 * hp0 + si01 * hp1;
        const float k1 = si01 * hp0 + si11 * hp1;

        float* sg = &sX[lb][0];

        // 6) mean update + predict (distributed element, gathered via scratch)
        const float mu = mmine + k0 * r0 + k1 * r1;
        sg[64 + j] = mu;
        LDSFENCE();
        float4 mua = *(const float4*)&sg[64], mub = *(const float4*)&sg[68];
        const float mean_u[8] = {mua.x, mua.y, mua.z, mua.w, mub.x, mub.y, mub.z, mub.w};
        float mp = 0.f;
        #pragma unroll
        for (int k = 0; k < 8; ++k) mp += Frow[k] * mean_u[k];
        LDSFENCE();
        sg[72 + j] = mp;
        LDSFENCE();
        float4 mpa = *(const float4*)&sg[72], mpb = *(const float4*)&sg[76];
        meanr[0] = mpa.x; meanr[1] = mpa.y; meanr[2] = mpa.z; meanr[3] = mpa.w;
        meanr[4] = mpb.x; meanr[5] = mpb.y; meanr[6] = mpb.z; meanr[7] = mpb.w;
        mmine = mp;
        means[((size_t)b * T_ + (t + 1)) * S_ + j] = mp;

        // 7) A = I - K H, row j
        float arow[8];
        #pragma unroll
        for (int k = 0; k < 8; ++k) arow[k] = idrow[k] - k0 * h0[k] - k1 * h1[k];

        // 8a) U = A P : column j = dot(A row i, pcol)
        LDSFENCE();
        *(float4*)&sg[j * 8]     = make_float4(arow[0], arow[1], arow[2], arow[3]);
        *(float4*)&sg[j * 8 + 4] = make_float4(arow[4], arow[5], arow[6], arow[7]);
        LDSFENCE();
        float u[8];
        #pragma unroll
        for (int i = 0; i < 8; ++i) {
            float4 ra = *(const float4*)&sg[i * 8], rb = *(const float4*)&sg[i * 8 + 4];
            u[i] = ra.x * pcol[0] + ra.y * pcol[1] + ra.z * pcol[2] + ra.w * pcol[3]
                 + rb.x * pcol[4] + rb.y * pcol[5] + rb.z * pcol[6] + rb.w * pcol[7];
        }
        // 8b) cov_u col j : C[i][j] = sum_k Ucol_k[i] * arow[k]
        LDSFENCE();
        *(float4*)&sg[j * 8]     = make_float4(u[0], u[1], u[2], u[3]);
        *(float4*)&sg[j * 8 + 4] = make_float4(u[4], u[5], u[6], u[7]);
        LDSFENCE();
        float cu[8] = {0.f, 0.f, 0.f, 0.f, 0.f, 0.f, 0.f, 0.f};
        #pragma unroll
        for (int k = 0; k < 8; ++k) {
            float4 ua = *(const float4*)&sg[k * 8], ub = *(const float4*)&sg[k * 8 + 4];
            const float ak = arow[k];
            cu[0] += ua.x * ak; cu[1] += ua.y * ak; cu[2] += ua.z * ak; cu[3] += ua.w * ak;
            cu[4] += ub.x * ak; cu[5] += ub.y * ak; cu[6] += ub.z * ak; cu[7] += ub.w * ak;
        }
        // 8c) + K R K^T col j
        const float w0 = R00 * k0 + R01 * k1;
        const float w1 = R01 * k0 + R11 * k1;
        LDSFENCE();
        sg[64 + j * 2] = k0; sg[64 + j * 2 + 1] = k1;
        LDSFENCE();
        {
            float4 kp0 = *(const float4*)&sg[64], kp1 = *(const float4*)&sg[68];
            float4 kp2 = *(const float4*)&sg[72], kp3 = *(const float4*)&sg[76];
            cu[0] += kp0.x * w0 + kp0.y * w1;  cu[1] += kp0.z * w0 + kp0.w * w1;
            cu[2] += kp1.x * w0 + kp1.y * w1;  cu[3] += kp1.z * w0 + kp1.w * w1;
            cu[4] += kp2.x * w0 + kp2.y * w1;  cu[5] += kp2.z * w0 + kp2.w * w1;
            cu[6] += kp3.x * w0 + kp3.y * w1;  cu[7] += kp3.z * w0 + kp3.w * w1;
        }

        // 9) predict: V = F cov_u (F rows straight from the TDM tile in LDS)
        float v[8];
        #pragma unroll
        for (int i = 0; i < 8; ++i) {
            float4 fa = f4[i * 2], fb = f4[i * 2 + 1];
            v[i] = fa.x * cu[0] + fa.y * cu[1] + fa.z * cu[2] + fa.w * cu[3]
                 + fb.x * cu[4] + fb.y * cu[5] + fb.z * cu[6] + fb.w * cu[7];
        }
        LDSFENCE();
        *(float4*)&sg[j * 8]     = make_float4(v[0], v[1], v[2], v[3]);
        *(float4*)&sg[j * 8 + 4] = make_float4(v[4], v[5], v[6], v[7]);
        LDSFENCE();
        float cp[8];
        #pragma unroll
        for (int i = 0; i < 8; ++i) cp[i] = qcol[i];
        #pragma unroll
        for (int k = 0; k < 8; ++k) {
            float4 va = *(const float4*)&sg[k * 8], vb = *(const float4*)&sg[k * 8 + 4];
            const float fk = Frow[k];
            cp[0] += va.x * fk; cp[1] += va.y * fk; cp[2] += va.z * fk; cp[3] += va.w * fk;
            cp[4] += vb.x * fk; cp[5] += vb.y * fk; cp[6] += vb.z * fk; cp[7] += vb.w * fk;
        }

        // commit + store cov row j (== column j by symmetry)
        #pragma unroll
        for (int k = 0; k < 8; ++k) pcol[k] = cp[k];
        float4* cd = (float4*)&covs[((size_t)b * T_ + (t + 1)) * 64 + j * 8];
        cd[0] = make_float4(cp[0], cp[1], cp[2], cp[3]);
        cd[1] = make_float4(cp[4], cp[5], cp[6], cp[7]);
    }
}

extern "C" void kernel_launch(void* const* d_in, const int* in_sizes, int n_in,
                              void* d_out, int out_size, void* d_ws, size_t ws_size,
                              hipStream_t stream) {
    (void)in_sizes; (void)n_in; (void)d_ws; (void)ws_size; (void)out_size;
    const float* y  = (const float*)d_in[0];
    const float* F  = (const float*)d_in[1];
    const float* H  = (const float*)d_in[2];
    const float* Q  = (const float*)d_in[3];
    const float* R  = (const float*)d_in[4];
    const float* im = (const float*)d_in[5];
    const float* ic = (const float*)d_in[6];
    // d_in[7] = n_step (==1), unused
    float* means = (float*)d_out;
    float* covs  = means + (size_t)B_ * T_ * S_;
    hipLaunchKernelGGL(kalman1250_kernel, dim3(NBLOCKS), dim3(THREADS), 0, stream,
                       y, F, H, Q, R, im, ic, means, covs);
}